// LaplacianLoss_5111011082240
// MI455X (gfx1250) — compile-verified
//
#include <hip/hip_runtime.h>

#define NV 16384
#define NB 4
#define KC 512                 // K-chunk of X^T staged in LDS (512*16*4B = 32 KB per buffer)
#define NCHUNK (NV / KC)       // 32
#define WAVES_PER_BLOCK 4
#define BLOCK (WAVES_PER_BLOCK * 32)
#define NTILES (NV / 16)       // 1024 independent 16-row tiles of L

typedef float v2f __attribute__((ext_vector_type(2)));
typedef float v8f __attribute__((ext_vector_type(8)));
typedef unsigned int u32x4 __attribute__((ext_vector_type(4)));
typedef int i32x4 __attribute__((ext_vector_type(4)));
typedef int i32x8 __attribute__((ext_vector_type(8)));

// Build padded transposed X: xt[j*16 + n] = x[b, j, d] with n = 3*b + d for n<12, else 0.
__global__ void prep_xt(const float* __restrict__ x, float* __restrict__ xt) {
    int idx = blockIdx.x * blockDim.x + threadIdx.x;   // 0 .. NV*16-1
    int j = idx >> 4;
    int n = idx & 15;
    float v = 0.0f;
    if (n < 12) {
        int b = n / 3, d = n % 3;
        v = x[(size_t)b * NV * 3 + (size_t)j * 3 + d];
    }
    xt[idx] = v;
}

// Issue one TDM DMA: contiguous 1D tile of KC*16 fp32 (32 KB) global -> LDS.
// D# built per CDNA5 ISA 8.3/8.4 (count=1, type=image(2), data_size=4B,
// tile_dim0 = tensor_dim0 = KC*16, tile_dim1 = 1).
__device__ __forceinline__ void tdm_load_chunk(const float* gsrc, unsigned lds_byte_addr) {
    unsigned long long ga = (unsigned long long)(uintptr_t)gsrc;
    u32x4 g0;
    g0.x = 1u;                                        // count=1, user mode, no gather
    g0.y = lds_byte_addr;                             // lds_addr [63:32]
    g0.z = (unsigned)(ga & 0xFFFFFFFFu);              // global_addr [95:64]
    g0.w = (unsigned)((ga >> 32) & 0x01FFFFFFu)       // global_addr [120:96]
           | 0x80000000u;                             // type=2 at [127:126]
    i32x8 g1;
    g1.s0 = 0x20000;                                  // data_size=2 (4 bytes); mask=0; no pad/iter
    g1.s1 = (int)((KC * 16) << 16);                   // tensor_dim0[15:0] at [63:48] (field bits 79:48)
    g1.s2 = (int)(((KC * 16) >> 16) | (1u << 16));    // tensor_dim0[31:16] | tensor_dim1[15:0]=1
    g1.s3 = (int)((unsigned)(KC * 16) << 16);         // tensor_dim1[31:16]=0 | tile_dim0 = KC*16
    g1.s4 = 1;                                        // tile_dim1 = 1, tile_dim2 = 0
    g1.s5 = KC * 16;                                  // tensor_dim0_stride (unused for 1 row)
    g1.s6 = 0;
    g1.s7 = 0;
    i32x4 gz = {0, 0, 0, 0};
#if defined(__clang_major__) && (__clang_major__ >= 23)
    i32x8 gz8 = {0, 0, 0, 0, 0, 0, 0, 0};
    __builtin_amdgcn_tensor_load_to_lds(g0, g1, gz, gz, gz8, 0);
#else
    __builtin_amdgcn_tensor_load_to_lds(g0, g1, gz, gz, 0);
#endif
}

// Each wave computes one 16-row tile of Y = L * X^T (N padded to 16) with
// V_WMMA_F32_16X16X4_F32; X^T chunks are double-buffered into LDS by the TDM.
__global__ __launch_bounds__(BLOCK) void lap_wmma(const float* __restrict__ L,
                                                  const float* __restrict__ xt,
                                                  float* __restrict__ part) {
    __shared__ float lds_xt[2][KC * 16];

    const int tid   = threadIdx.x;
    const int wave  = tid >> 5;
    const int lane  = tid & 31;
    const int hi    = lane >> 4;       // K half: lanes 0-15 hold K=0,1; 16-31 hold K=2,3
    const int mlane = lane & 15;       // row within tile (same row in both lane halves)
    const int tile  = blockIdx.x * WAVES_PER_BLOCK + wave;
    const int row   = tile * 16 + mlane;
    const float* Lrow = L + (size_t)row * NV;

    v8f c = {0.f, 0.f, 0.f, 0.f, 0.f, 0.f, 0.f, 0.f};

    // Prologue: DMA chunk 0 into buffer 0 (wave 0 only; TDM ignores EXEC, one issue per wave)
    if (wave == 0) {
        tdm_load_chunk(xt, (unsigned)(uintptr_t)&lds_xt[0][0]);
    }

    for (int ch = 0; ch < NCHUNK; ++ch) {
        const int k_base = ch * KC;
        if (wave == 0) {
            if (ch + 1 < NCHUNK) {
                // overlap next chunk's DMA with this chunk's WMMAs
                tdm_load_chunk(xt + (size_t)(k_base + KC) * 16,
                               (unsigned)(uintptr_t)&lds_xt[(ch + 1) & 1][0]);
                __builtin_amdgcn_s_wait_tensorcnt(1);  // oldest (chunk ch) complete
            } else {
                __builtin_amdgcn_s_wait_tensorcnt(0);
            }
        }
        __syncthreads();   // publish buffer ch&1 to all waves

        // Per-chunk base pointers -> all inner DS/VMEM offsets become immediates
        const float* lb = &lds_xt[ch & 1][32 * hi + mlane];   // B fragment base
        const float* La = Lrow + k_base + 2 * hi;             // A fragment base

        for (int kk = 0; kk < KC; kk += 64) {
            __builtin_prefetch(La + kk + 2048, 0, 1);  // stream L ahead (speculative, OOB-safe)
            const float* lbk = lb + kk * 16;
            const float* Lak = La + kk;
#pragma unroll
            for (int t = 0; t < 16; ++t) {
                // A fragment: 16x4 fp32 tile of L; lane holds (row, K = 4t + 2*hi .. +1)
                v2f a = *(const v2f*)(Lak + 4 * t);
                // B fragment: 4x16 fp32 tile of X^T from LDS, immediate offsets
                v2f b;
                b.x = lbk[64 * t];
                b.y = lbk[64 * t + 16];
                c = __builtin_amdgcn_wmma_f32_16x16x4_f32(
                        /*neg_a=*/false, a, /*neg_b=*/false, b,
                        /*c_mod=*/(short)0, c, /*reuse_a=*/false, /*reuse_b=*/false);
            }
        }
        __syncthreads();   // all waves done reading buffer ch&1 before it is re-filled
    }

    // Per-lane sum of squares over its 8 C elements (pad columns n>=12 are exact zeros)
    float s = 0.0f;
#pragma unroll
    for (int v = 0; v < 8; ++v) s += c[v] * c[v];
    part[(size_t)tile * 32 + lane] = s;
}

// Deterministic reduction: out[b] = sum over tiles, halves, d of partials in column 3b+d.
__global__ void reduce_loss(const float* __restrict__ part, float* __restrict__ out) {
    __shared__ float red[256];
    const int b = blockIdx.x;       // 0..3
    const int t = threadIdx.x;      // 0..255
    float acc = 0.0f;
    for (int idx = t; idx < NTILES * 6; idx += 256) {
        int tile = idx / 6;
        int q    = idx % 6;
        int half = q / 3;
        int d    = q % 3;
        int lane = half * 16 + b * 3 + d;
        acc += part[(size_t)tile * 32 + lane];
    }
    red[t] = acc;
    __syncthreads();
    for (int ofs = 128; ofs > 0; ofs >>= 1) {
        if (t < ofs) red[t] += red[t + ofs];
        __syncthreads();
    }
    if (t == 0) out[b] = red[0];
}

extern "C" void kernel_launch(void* const* d_in, const int* in_sizes, int n_in,
                              void* d_out, int out_size, void* d_ws, size_t ws_size,
                              hipStream_t stream) {
    const float* x = (const float*)d_in[0];   // [B, NV, 3] fp32
    const float* L = (const float*)d_in[1];   // [NV, NV]   fp32
    float* out = (float*)d_out;               // [B] fp32

    float* xt   = (float*)d_ws;                                   // NV*16 floats (1 MB)
    float* part = (float*)((char*)d_ws + (size_t)NV * 16 * 4);    // NTILES*32 floats (128 KB)

    // 1) build padded transposed X
    prep_xt<<<(NV * 16) / 256, 256, 0, stream>>>(x, xt);
    // 2) WMMA main pass: 1024 tiles, 4 waves per block, TDM double-buffered X^T
    lap_wmma<<<NTILES / WAVES_PER_BLOCK, BLOCK, 0, stream>>>(L, xt, part);
    // 3) deterministic per-batch reduction
    reduce_loss<<<NB, 256, 0, stream>>>(part, out);
}